// GNNDemo_50113678409912
// MI455X (gfx1250) — compile-verified
//
#include <hip/hip_runtime.h>

// ---------------------------------------------------------------------------
// GCN (3x GCNConv + linear head) for MI455X / gfx1250, wave32 + WMMA fp32.
//   N=50000 nodes, E=800000 edges, IN=64, hidden 16/32, OUT=64.
// Strategy:
//   * dinv / per-edge norm computed ONCE (identical across the 3 layers).
//   * node-feature GEMMs via V_WMMA_F32_16X16X4_F32, one 16x16 tile per wave.
//   * edge aggregation via native global_atomic_add_f32 (relaxed, agent scope
//     so the compiler emits the L2 near-atomic instead of a CAS loop).
//   * self-loop term + bias + ReLU fused into one epilogue kernel.
// ---------------------------------------------------------------------------

typedef float v2f __attribute__((ext_vector_type(2)));
typedef float v8f __attribute__((ext_vector_type(8)));

#define GCN_N 50000
#define GCN_E 800000

// Native fp32 atomic add (agent scope, relaxed) -> global_atomic_add_f32.
__device__ __forceinline__ void fadd_atomic(float* p, float v) {
  __hip_atomic_fetch_add(p, v, __ATOMIC_RELAXED, __HIP_MEMORY_SCOPE_AGENT);
}

// ---------------- utility kernels ----------------

__global__ void fill_kernel(float* __restrict__ p, float v, int n) {
  int i = blockIdx.x * blockDim.x + threadIdx.x;
  if (i < n) p[i] = v;
}

// deg[col[e]] += ew[e]   (deg pre-initialized to 1.0 == self-loop weight)
__global__ void deg_kernel(const int* __restrict__ col, const float* __restrict__ ew,
                           float* __restrict__ deg, int e) {
  int i = blockIdx.x * blockDim.x + threadIdx.x;
  if (i < e) fadd_atomic(&deg[col[i]], ew[i]);
}

// deg -> dinv in place
__global__ void dinv_kernel(float* __restrict__ deg, int n) {
  int i = blockIdx.x * blockDim.x + threadIdx.x;
  if (i < n) {
    float d = deg[i];
    deg[i] = (d > 0.0f) ? rsqrtf(d) : 0.0f;
  }
}

// norm[e] = dinv[row[e]] * ew[e] * dinv[col[e]]
__global__ void norm_kernel(const int* __restrict__ row, const int* __restrict__ col,
                            const float* __restrict__ ew, const float* __restrict__ dinv,
                            float* __restrict__ nrm, int e) {
  int i = blockIdx.x * blockDim.x + threadIdx.x;
  if (i < e) nrm[i] = dinv[row[i]] * ew[i] * dinv[col[i]];
}

// ---------------- WMMA GEMM: H[M,F] = A[M,K] @ W[K,F] ----------------
// One wave computes one 16x16 output tile using V_WMMA_F32_16X16X4_F32.
// A layout (16x4 fp32): lanes 0-15 hold K = k0+{0,1}; lanes 16-31 K = k0+{2,3}.
// B layout (4x16 fp32): vgpr0 = rows k0 / k0+2, vgpr1 = rows k0+1 / k0+3.
// C/D: vgpr v -> row v + 8*(lane>=16), col = lane&15.
template <int K, int F>
__global__ void gemm_wmma_kernel(const float* __restrict__ A, const float* __restrict__ W,
                                 float* __restrict__ H, int M) {
  const int wave = (blockIdx.x * blockDim.x + threadIdx.x) >> 5;
  const int lane = threadIdx.x & 31;
  const int tiles_n = F / 16;
  const int tiles = (M / 16) * tiles_n;
  if (wave >= tiles) return;
  const int tm = wave / tiles_n;
  const int tn = wave % tiles_n;
  const int half = lane >> 4;  // 0: lanes 0-15, 1: lanes 16-31
  const int l = lane & 15;

  v8f c = {};
  const float* arow = A + (size_t)(tm * 16 + l) * K;
  const int ncol = tn * 16 + l;
#pragma unroll
  for (int k0 = 0; k0 < K; k0 += 4) {
    v2f a, b;
    a.x = arow[k0 + 2 * half + 0];
    a.y = arow[k0 + 2 * half + 1];
    b.x = W[(size_t)(k0 + 2 * half + 0) * F + ncol];
    b.y = W[(size_t)(k0 + 2 * half + 1) * F + ncol];
    c = __builtin_amdgcn_wmma_f32_16x16x4_f32(false, a, false, b, (short)0, c,
                                              false, false);
  }
  float* o = H + (size_t)(tm * 16) * F + ncol;
#pragma unroll
  for (int v = 0; v < 8; ++v) {
    o[(size_t)(v + 8 * half) * F] = c[v];
  }
}

// ---------------- final GEMM: OUT = [X | X4] @ W + bias  (K=128, F=64) -------
__global__ void final_gemm_kernel(const float* __restrict__ X, const float* __restrict__ X4,
                                  const float* __restrict__ W, const float* __restrict__ bias,
                                  float* __restrict__ OUT, int M) {
  const int F = 64;
  const int wave = (blockIdx.x * blockDim.x + threadIdx.x) >> 5;
  const int lane = threadIdx.x & 31;
  const int tiles_n = F / 16;
  const int tiles = (M / 16) * tiles_n;
  if (wave >= tiles) return;
  const int tm = wave / tiles_n;
  const int tn = wave % tiles_n;
  const int half = lane >> 4;
  const int l = lane & 15;
  const int ncol = tn * 16 + l;

  v8f c = {};
  const float* xrow  = X  + (size_t)(tm * 16 + l) * 64;
  const float* x4row = X4 + (size_t)(tm * 16 + l) * 64;
#pragma unroll
  for (int k0 = 0; k0 < 128; k0 += 4) {
    const float* arow = (k0 < 64) ? (xrow + k0) : (x4row + (k0 - 64));
    v2f a, b;
    a.x = arow[2 * half + 0];
    a.y = arow[2 * half + 1];
    b.x = W[(size_t)(k0 + 2 * half + 0) * F + ncol];
    b.y = W[(size_t)(k0 + 2 * half + 1) * F + ncol];
    c = __builtin_amdgcn_wmma_f32_16x16x4_f32(false, a, false, b, (short)0, c,
                                              false, false);
  }
  const float bb = bias[ncol];
  float* o = OUT + (size_t)(tm * 16) * F + ncol;
#pragma unroll
  for (int v = 0; v < 8; ++v) {
    o[(size_t)(v + 8 * half) * F] = c[v] + bb;
  }
}

// ---------------- edge scatter: OUT[col[e],:] += norm[e] * H[row[e],:] -------
// grid.x over edges, grid.y over feature groups of 4 (float4 gather, 4 atomics)
template <int F>
__global__ void scatter_kernel(const int* __restrict__ row, const int* __restrict__ col,
                               const float* __restrict__ nrm, const float* __restrict__ H,
                               float* __restrict__ OUT, int e) {
  int edge = blockIdx.x * blockDim.x + threadIdx.x;
  if (edge >= e) return;
  const int g = blockIdx.y * 4;
  const float nm = nrm[edge];
  const float4 h = *(const float4*)(H + (size_t)row[edge] * F + g);
  float* o = OUT + (size_t)col[edge] * F + g;
  fadd_atomic(o + 0, nm * h.x);
  fadd_atomic(o + 1, nm * h.y);
  fadd_atomic(o + 2, nm * h.z);
  fadd_atomic(o + 3, nm * h.w);
}

// ---------------- epilogue: ACC = relu(ACC + dinv^2 * H + b) -----------------
template <int F>
__global__ void finalize_kernel(float* __restrict__ ACC, const float* __restrict__ H,
                                const float* __restrict__ dinv, const float* __restrict__ b,
                                int n) {
  int t = blockIdx.x * blockDim.x + threadIdx.x;
  if (t >= n * F) return;
  const int i = t / F;
  const int f = t % F;
  const float di = dinv[i];
  float v = ACC[t] + di * di * H[t] + b[f];
  ACC[t] = (v > 0.0f) ? v : 0.0f;
}

// ---------------------------------------------------------------------------

static inline int cdiv(int a, int b) { return (a + b - 1) / b; }

extern "C" void kernel_launch(void* const* d_in, const int* in_sizes, int n_in,
                              void* d_out, int out_size, void* d_ws, size_t ws_size,
                              hipStream_t stream) {
  const float* x  = (const float*)d_in[0];   // [N,64]
  const int*   ei = (const int*)d_in[1];     // [2,E]
  const float* ea = (const float*)d_in[2];   // [E]
  const float* w1 = (const float*)d_in[3];   // [64,16]
  const float* b1 = (const float*)d_in[4];   // [16]
  const float* w2 = (const float*)d_in[5];   // [16,32]
  const float* b2 = (const float*)d_in[6];   // [32]
  const float* w3 = (const float*)d_in[7];   // [32,64]
  const float* b3 = (const float*)d_in[8];   // [64]
  const float* lw = (const float*)d_in[9];   // [128,64]
  const float* lb = (const float*)d_in[10];  // [64]
  float* out = (float*)d_out;                // [N,64]

  const int N = GCN_N, E = GCN_E;
  const int* row = ei;
  const int* col = ei + E;

  // workspace layout (floats): dinv[N] | norm[E] | bufH[N*64] | bufX[N*64]
  float* ws   = (float*)d_ws;
  float* dinv = ws;
  float* nrm  = ws + N;
  float* bufH = nrm + E;
  float* bufX = bufH + (size_t)N * 64;

  // ---- normalization (computed once; identical for all 3 layers) ----
  fill_kernel<<<cdiv(N, 256), 256, 0, stream>>>(dinv, 1.0f, N);  // self-loop weight
  deg_kernel<<<cdiv(E, 256), 256, 0, stream>>>(col, ea, dinv, E);
  dinv_kernel<<<cdiv(N, 256), 256, 0, stream>>>(dinv, N);
  norm_kernel<<<cdiv(E, 256), 256, 0, stream>>>(row, col, ea, dinv, nrm, E);

  const int tiles_m = N / 16;  // 3125, exact

  // ---- layer 1: [N,64] @ [64,16] ----
  {
    const int F = 16;
    gemm_wmma_kernel<64, 16><<<cdiv(tiles_m * (F / 16), 8), 256, 0, stream>>>(x, w1, bufH, N);
    fill_kernel<<<cdiv(N * F, 256), 256, 0, stream>>>(bufX, 0.0f, N * F);
    dim3 g(cdiv(E, 256), F / 4);
    scatter_kernel<16><<<g, 256, 0, stream>>>(row, col, nrm, bufH, bufX, E);
    finalize_kernel<16><<<cdiv(N * F, 256), 256, 0, stream>>>(bufX, bufH, dinv, b1, N);
  }
  // ---- layer 2: [N,16] @ [16,32] ----
  {
    const int F = 32;
    gemm_wmma_kernel<16, 32><<<cdiv(tiles_m * (F / 16), 8), 256, 0, stream>>>(bufX, w2, bufH, N);
    fill_kernel<<<cdiv(N * F, 256), 256, 0, stream>>>(bufX, 0.0f, N * F);
    dim3 g(cdiv(E, 256), F / 4);
    scatter_kernel<32><<<g, 256, 0, stream>>>(row, col, nrm, bufH, bufX, E);
    finalize_kernel<32><<<cdiv(N * F, 256), 256, 0, stream>>>(bufX, bufH, dinv, b2, N);
  }
  // ---- layer 3: [N,32] @ [32,64] ----
  {
    const int F = 64;
    gemm_wmma_kernel<32, 64><<<cdiv(tiles_m * (F / 16), 8), 256, 0, stream>>>(bufX, w3, bufH, N);
    fill_kernel<<<cdiv(N * F, 256), 256, 0, stream>>>(bufX, 0.0f, N * F);
    dim3 g(cdiv(E, 256), F / 4);
    scatter_kernel<64><<<g, 256, 0, stream>>>(row, col, nrm, bufH, bufX, E);
    finalize_kernel<64><<<cdiv(N * F, 256), 256, 0, stream>>>(bufX, bufH, dinv, b3, N);
  }
  // ---- head: [x | x4] @ [128,64] + lb ----
  final_gemm_kernel<<<cdiv(tiles_m * 4, 8), 256, 0, stream>>>(x, bufX, lw, lb, out, N);
}